// HierarchicalLoss_77386720740083
// MI455X (gfx1250) — compile-verified
//
#include <hip/hip_runtime.h>
#include <hip/hip_bf16.h>
#include <math.h>
#include <stdint.h>

typedef __attribute__((ext_vector_type(2))) float v2f;
typedef __attribute__((ext_vector_type(8))) float v8f;

#define INV_TEMP (1.0f / 0.07f)
#define IGNORE_IDX (-100)

// ---------------------------------------------------------------------------
// Online softmax helpers
// ---------------------------------------------------------------------------
__device__ __forceinline__ void online_update(float x, float& m, float& s) {
    float nm = fmaxf(m, x);
    s = s * __expf(m - nm) + __expf(x - nm);
    m = nm;
}

// Chunked update: one rescale + 4 exps per 4 elements (1.25 exp/elem).
__device__ __forceinline__ void online_update4(float4 v, float& m, float& s) {
    float cmax = fmaxf(fmaxf(v.x, v.y), fmaxf(v.z, v.w));
    float nm   = fmaxf(m, cmax);
    s = s * __expf(m - nm)
      + __expf(v.x - nm) + __expf(v.y - nm)
      + __expf(v.z - nm) + __expf(v.w - nm);
    m = nm;
}

__device__ __forceinline__ void merge_ms(float m2, float s2, float& m, float& s) {
    float M = fmaxf(m, m2);
    float S = (m  == -INFINITY ? 0.0f : s  * __expf(m  - M)) +
              (m2 == -INFINITY ? 0.0f : s2 * __expf(m2 - M));
    m = M; s = S;
}

// ---------------------------------------------------------------------------
// 1) MLM: one block per row, single-pass online softmax over V=30522.
//    Rows are only 8B-aligned in general, so: scalar prologue (0 or 2 floats)
//    to a 16B boundary, aligned float4 bulk (global_load_b128), scalar tail.
// ---------------------------------------------------------------------------
__global__ void mlm_row_kernel(const float* __restrict__ logits,
                               const int*   __restrict__ labels,
                               float* __restrict__ nll_out,
                               float* __restrict__ valid_out,
                               int V) {
    const int row = blockIdx.x;
    const int t   = threadIdx.x;
    const int NT  = blockDim.x;
    const float* rp = logits + (size_t)row * (size_t)V;

    // floats to reach a 16B boundary: base is always 8B aligned -> 0 or 2
    const int pre  = (int)(((16u - ((uint32_t)(uintptr_t)rp & 15u)) & 15u) >> 2);
    const int nvec = (V - pre) >> 2;          // aligned float4 chunks
    const int tail = pre + (nvec << 2);       // first scalar-tail index

    float m = -INFINITY, s = 0.0f;

    for (int i = t; i < pre; i += NT) online_update(rp[i], m, s);

    const float4* rp4 = reinterpret_cast<const float4*>(rp + pre);
    for (int i = t; i < nvec; i += NT) online_update4(rp4[i], m, s);

    for (int i = tail + t; i < V; i += NT) online_update(rp[i], m, s);

    __shared__ float sm[256];
    __shared__ float ss[256];
    sm[t] = m; ss[t] = s;
    __syncthreads();
    for (int off = NT >> 1; off > 0; off >>= 1) {
        if (t < off) {
            float mm = sm[t], sv = ss[t];
            merge_ms(sm[t + off], ss[t + off], mm, sv);
            sm[t] = mm; ss[t] = sv;
        }
        __syncthreads();
    }

    if (t == 0) {
        int lab = labels[row];
        bool valid = (lab != IGNORE_IDX);
        float lse = sm[0] + __logf(ss[0]);
        float xl  = valid ? rp[lab] : 0.0f;
        nll_out[row]   = valid ? (lse - xl) : 0.0f;
        valid_out[row] = valid ? 1.0f : 0.0f;
    }
}

// ---------------------------------------------------------------------------
// 2) Row-wise L2 normalize (D multiple of 4, 16B-aligned rows).
// ---------------------------------------------------------------------------
__global__ void l2norm_kernel(const float* __restrict__ in,
                              float* __restrict__ out, int D) {
    const int row = blockIdx.x;
    const int t   = threadIdx.x;
    const int D4  = D >> 2;
    const float4* ip = reinterpret_cast<const float4*>(in  + (size_t)row * D);
    float4*       op = reinterpret_cast<float4*>      (out + (size_t)row * D);

    float acc = 0.0f;
    for (int i = t; i < D4; i += blockDim.x) {
        float4 v = ip[i];
        acc += v.x * v.x + v.y * v.y + v.z * v.z + v.w * v.w;
    }

    __shared__ float sh[256];
    sh[t] = acc; __syncthreads();
    for (int off = blockDim.x >> 1; off > 0; off >>= 1) {
        if (t < off) sh[t] += sh[t + off];
        __syncthreads();
    }
    float inv = 1.0f / fmaxf(sqrtf(sh[0]), 1e-12f);
    for (int i = t; i < D4; i += blockDim.x) {
        float4 v = ip[i];
        v.x *= inv; v.y *= inv; v.z *= inv; v.w *= inv;
        op[i] = v;
    }
}

// ---------------------------------------------------------------------------
// 3) pos_sim[p] = dot(a_n[p], p_n[p]) / TEMP, one block per row.
// ---------------------------------------------------------------------------
__global__ void pos_sim_kernel(const float* __restrict__ a,
                               const float* __restrict__ p,
                               float* __restrict__ out, int D) {
    const int row = blockIdx.x;
    const int t   = threadIdx.x;
    const int D4  = D >> 2;
    const float4* ap = reinterpret_cast<const float4*>(a + (size_t)row * D);
    const float4* pp = reinterpret_cast<const float4*>(p + (size_t)row * D);
    float acc = 0.0f;
    for (int i = t; i < D4; i += blockDim.x) {
        float4 va = ap[i], vp = pp[i];
        acc += va.x * vp.x + va.y * vp.y + va.z * vp.z + va.w * vp.w;
    }
    __shared__ float sh[256];
    sh[t] = acc; __syncthreads();
    for (int off = blockDim.x >> 1; off > 0; off >>= 1) {
        if (t < off) sh[t] += sh[t + off];
        __syncthreads();
    }
    if (t == 0) out[row] = sh[0] * INV_TEMP;
}

// ---------------------------------------------------------------------------
// 4) S = (A @ B^T) * scale using V_WMMA_F32_16X16X4_F32.
//    A: [M, D] row-major, B: [N, D] row-major, S: [M, N]; M, N multiples of 16.
//    One wave per 16x16 output tile; K-loop over D in steps of 4.
//    Per ISA 7.12.2 (32-bit A 16x4 / B 4x16): lane<16 holds {K,K+1} of row
//    (lane&15); lane>=16 holds {K+2,K+3}. Since B operand is N^T, both A and
//    B lanes fetch 2 contiguous floats from a row-major row. C/D: acc[v] is
//    row tm+v+8*(lane>=16), col tn+(lane&15).
// ---------------------------------------------------------------------------
__global__ void wmma_gemm_nt_kernel(const float* __restrict__ A,
                                    const float* __restrict__ B,
                                    float* __restrict__ S,
                                    int M, int N, int D, float scale) {
    const int wave = (int)((blockIdx.x * blockDim.x + threadIdx.x) >> 5);
    const int lane = threadIdx.x & 31;
    const int tilesN = N >> 4;
    const int ntiles = (M >> 4) * tilesN;
    if (wave >= ntiles) return;  // wave-uniform: EXEC stays all-ones for WMMA

    const int tm = (wave / tilesN) << 4;
    const int tn = (wave % tilesN) << 4;
    const int r    = lane & 15;
    const int koff = (lane >> 4) << 1;      // 0 or 2

    const float* ap = A + (size_t)(tm + r) * D + koff;
    const float* bp = B + (size_t)(tn + r) * D + koff;

    v8f acc = {};
    for (int k = 0; k < D; k += 4) {
        v2f av = *reinterpret_cast<const v2f*>(ap + k);
        v2f bv = *reinterpret_cast<const v2f*>(bp + k);
        acc = __builtin_amdgcn_wmma_f32_16x16x4_f32(
            /*neg_a=*/false, av, /*neg_b=*/false, bv,
            /*c_mod=*/(short)0, acc, /*reuse_a=*/false, /*reuse_b=*/false);
    }

    const int rhalf = (lane >> 4) << 3;     // 0 or 8
    const int col   = tn + (lane & 15);
#pragma unroll
    for (int v = 0; v < 8; ++v) {
        S[(size_t)(tm + rhalf + v) * N + col] = acc[v] * scale;
    }
}

// ---------------------------------------------------------------------------
// 5) InfoNCE row loss: loss[p] = logsumexp([pos[p], neg[p,:]]) - pos[p]
// ---------------------------------------------------------------------------
__global__ void infonce_row_kernel(const float* __restrict__ pos,
                                   const float* __restrict__ neg,
                                   float* __restrict__ loss, int N) {
    const int p = blockIdx.x;
    const int t = threadIdx.x;
    const float* np = neg + (size_t)p * N;

    float m = -INFINITY, s = 0.0f;
    for (int i = t; i < N; i += blockDim.x) online_update(np[i], m, s);

    __shared__ float sm[256];
    __shared__ float ss[256];
    sm[t] = m; ss[t] = s;
    __syncthreads();
    for (int off = blockDim.x >> 1; off > 0; off >>= 1) {
        if (t < off) {
            float mm = sm[t], sv = ss[t];
            merge_ms(sm[t + off], ss[t + off], mm, sv);
            sm[t] = mm; ss[t] = sv;
        }
        __syncthreads();
    }
    if (t == 0) {
        float ps = pos[p];
        float M = fmaxf(sm[0], ps);
        float S = ss[0] * __expf(sm[0] - M) + __expf(ps - M);
        loss[p] = M + __logf(S) - ps;
    }
}

// ---------------------------------------------------------------------------
// 6) Sonnet row loss from 16x16 sim matrix (already scaled by 1/TEMP).
// ---------------------------------------------------------------------------
__global__ void sonnet_row_kernel(const float* __restrict__ sim,
                                  float* __restrict__ loss, int B) {
    const int r = threadIdx.x;
    if (r < B) {
        const float* rp = sim + (size_t)r * B;
        float m = -INFINITY;
        for (int j = 0; j < B; ++j) m = fmaxf(m, rp[j]);
        float s = 0.0f;
        for (int j = 0; j < B; ++j) s += __expf(rp[j] - m);
        loss[r] = m + __logf(s) - rp[r];
    }
}

// ---------------------------------------------------------------------------
// 7) Final deterministic combine -> d_out[0]
// ---------------------------------------------------------------------------
__global__ void finalize_kernel(const float* __restrict__ mlm_nll,
                                const float* __restrict__ mlm_val, int NR,
                                const float* __restrict__ line_loss, int PL,
                                const float* __restrict__ quat_loss, int PQ,
                                const float* __restrict__ son_loss,  int BS,
                                float* __restrict__ out) {
    __shared__ float sh[256];
    const int t = threadIdx.x;

    auto block_sum = [&](const float* a, int n) -> float {
        float s = 0.0f;
        for (int i = t; i < n; i += 256) s += a[i];
        sh[t] = s; __syncthreads();
        for (int off = 128; off > 0; off >>= 1) {
            if (t < off) sh[t] += sh[t + off];
            __syncthreads();
        }
        float r = sh[0];
        __syncthreads();
        return r;
    };

    float nll  = block_sum(mlm_nll, NR);
    float cnt  = block_sum(mlm_val, NR);
    float line = block_sum(line_loss, PL) / (float)PL;
    float quat = block_sum(quat_loss, PQ) / (float)PQ;
    float son  = block_sum(son_loss,  BS) / (float)BS;

    if (t == 0) {
        float mlm = nll / fmaxf(cnt, 1.0f);
        out[0] = 0.5f * mlm + 0.2f * line + 0.2f * quat + 0.1f * son;
    }
}

// ---------------------------------------------------------------------------
// Host launcher
// ---------------------------------------------------------------------------
extern "C" void kernel_launch(void* const* d_in, const int* in_sizes, int n_in,
                              void* d_out, int out_size, void* d_ws, size_t ws_size,
                              hipStream_t stream) {
    (void)in_sizes; (void)n_in; (void)out_size; (void)ws_size;

    constexpr int B = 16, Seq = 256, V = 30522, D = 768;
    constexpr int NR = B * Seq;           // 4096 MLM rows
    constexpr int PL = 224, NL = 224;     // line
    constexpr int PQ = 64,  NQ = 64;      // quat

    const float* mlm_logits = (const float*)d_in[0];
    const int*   mlm_labels = (const int*)  d_in[1];
    const float* line_a = (const float*)d_in[2];
    const float* line_p = (const float*)d_in[3];
    const float* line_n = (const float*)d_in[4];
    const float* quat_a = (const float*)d_in[5];
    const float* quat_p = (const float*)d_in[6];
    const float* quat_n = (const float*)d_in[7];
    const float* son_e  = (const float*)d_in[8];
    float* out = (float*)d_out;

    // Workspace carve-out (~2.96 MB of floats; all offsets 16B-aligned)
    float* w = (float*)d_ws;
    float* mlm_nll = w;  w += NR;
    float* mlm_val = w;  w += NR;
    float* la = w;       w += PL * D;
    float* lp = w;       w += PL * D;
    float* ln = w;       w += NL * D;
    float* lpos = w;     w += PL;
    float* lneg = w;     w += PL * NL;
    float* lloss = w;    w += PL;
    float* qa = w;       w += PQ * D;
    float* qp = w;       w += PQ * D;
    float* qn = w;       w += NQ * D;
    float* qpos = w;     w += PQ;
    float* qneg = w;     w += PQ * NQ;
    float* qloss = w;    w += PQ;
    float* sn = w;       w += B * D;
    float* ssim = w;     w += B * B;
    float* sloss = w;    w += B;

    // --- MLM (the bandwidth-dominant part: one b128 pass over 500 MB) ---
    mlm_row_kernel<<<NR, 256, 0, stream>>>(mlm_logits, mlm_labels, mlm_nll, mlm_val, V);

    // --- Normalize all contrastive inputs ---
    l2norm_kernel<<<PL, 256, 0, stream>>>(line_a, la, D);
    l2norm_kernel<<<PL, 256, 0, stream>>>(line_p, lp, D);
    l2norm_kernel<<<NL, 256, 0, stream>>>(line_n, ln, D);
    l2norm_kernel<<<PQ, 256, 0, stream>>>(quat_a, qa, D);
    l2norm_kernel<<<PQ, 256, 0, stream>>>(quat_p, qp, D);
    l2norm_kernel<<<NQ, 256, 0, stream>>>(quat_n, qn, D);
    l2norm_kernel<<<B,  256, 0, stream>>>(son_e,  sn, D);

    // --- Positive similarities ---
    pos_sim_kernel<<<PL, 256, 0, stream>>>(la, lp, lpos, D);
    pos_sim_kernel<<<PQ, 256, 0, stream>>>(qa, qp, qpos, D);

    // --- Similarity GEMMs on the WMMA pipe (fp32, 16x16x4) ---
    {
        int tiles = (PL / 16) * (NL / 16);            // 196
        int blocks = (tiles + 7) / 8;                 // 8 waves / 256-thr block
        wmma_gemm_nt_kernel<<<blocks, 256, 0, stream>>>(la, ln, lneg, PL, NL, D, INV_TEMP);
    }
    {
        int tiles = (PQ / 16) * (NQ / 16);            // 16
        int blocks = (tiles + 7) / 8;
        wmma_gemm_nt_kernel<<<blocks, 256, 0, stream>>>(qa, qn, qneg, PQ, NQ, D, INV_TEMP);
    }
    wmma_gemm_nt_kernel<<<1, 32, 0, stream>>>(sn, sn, ssim, B, B, D, INV_TEMP);

    // --- Row losses ---
    infonce_row_kernel<<<PL, 256, 0, stream>>>(lpos, lneg, lloss, NL);
    infonce_row_kernel<<<PQ, 256, 0, stream>>>(qpos, qneg, qloss, NQ);
    sonnet_row_kernel<<<1, 32, 0, stream>>>(ssim, sloss, B);

    // --- Weighted total ---
    finalize_kernel<<<1, 256, 0, stream>>>(mlm_nll, mlm_val, NR,
                                           lloss, PL, qloss, PQ, sloss, B, out);
}